// Former_25872882991174
// MI455X (gfx1250) — compile-verified
//
#include <hip/hip_runtime.h>
#include <hip/hip_bf16.h>

typedef __attribute__((ext_vector_type(2))) float v2f;
typedef __attribute__((ext_vector_type(8))) float v8f;

#define HF 96
#define WF 192
#define CF 32
#define DFULL 192
#define DC 48
#define HO 384
#define WO 768
#define LDST 196            // padded LDS row stride (floats) to avoid bank conflicts
#define LDS_BYTES (WF * LDST * 4)

// ---------------------------------------------------------------------------
// Kernel 1: channel-wise cosine normalization, relayout to [h][c][w]
// ---------------------------------------------------------------------------
__global__ __launch_bounds__(256) void normalize_feats(const float* __restrict__ lf,
                                                       const float* __restrict__ rf,
                                                       float* __restrict__ rn,
                                                       float* __restrict__ tn) {
  int pix = blockIdx.x * blockDim.x + threadIdx.x;
  if (pix >= HF * WF) return;
  int h = pix / WF;
  int w = pix - h * WF;

  float a[CF], b[CF];
  float ss0 = 0.f, ss1 = 0.f;
#pragma unroll
  for (int c = 0; c < CF; ++c) {
    a[c] = lf[(size_t)c * (HF * WF) + pix];
    b[c] = rf[(size_t)c * (HF * WF) + pix];
    ss0 += a[c] * a[c];
    ss1 += b[c] * b[c];
  }
  float i0 = rsqrtf(ss0 + 1e-12f);
  float i1 = rsqrtf(ss1 + 1e-12f);
  size_t base = (size_t)h * (CF * WF) + w;
#pragma unroll
  for (int c = 0; c < CF; ++c) {
    rn[base + (size_t)c * WF] = a[c] * i0;
    tn[base + (size_t)c * WF] = b[c] * i1;
  }
}

// ---------------------------------------------------------------------------
// Kernel 2: per-row 192x192x32 correlation GEMM via V_WMMA_F32_16X16X4_F32,
// staged through LDS, then diagonal-scatter to cost volume (coalesced stores).
// 12 waves/block (wave32); wave wv owns rows [16*wv, 16*wv+16) of M.
// ---------------------------------------------------------------------------
__global__ __launch_bounds__(384) void corr_rows(const float* __restrict__ rn,
                                                 const float* __restrict__ tn,
                                                 float* __restrict__ cost) {
  extern __shared__ float M[];  // WF * LDST floats (~147 KB, CDNA5 WGP LDS)
  const int h = blockIdx.x;
  const float* __restrict__ rrow = rn + (size_t)h * (CF * WF);
  const float* __restrict__ trow = tn + (size_t)h * (CF * WF);

  const int wave = threadIdx.x >> 5;   // 0..11
  const int lane = threadIdx.x & 31;
  const int half = lane >> 4;          // 0: lanes 0-15, 1: lanes 16-31
  const int l    = lane & 15;
  const int m0   = wave * 16;

  // Preload A fragments (16x4 fp32 tiles): lane<16 holds K=0,1; lane>=16 holds K=2,3
  v2f afr[8];
#pragma unroll
  for (int ks = 0; ks < 8; ++ks) {
    const int c0 = ks * 4 + half * 2;
    afr[ks].x = rrow[(c0 + 0) * WF + m0 + l];
    afr[ks].y = rrow[(c0 + 1) * WF + m0 + l];
  }

#pragma unroll
  for (int nt = 0; nt < 12; ++nt) {
    const int n0 = nt * 16;
    v8f acc = {0.f, 0.f, 0.f, 0.f, 0.f, 0.f, 0.f, 0.f};
#pragma unroll
    for (int ks = 0; ks < 8; ++ks) {
      const int c0 = ks * 4 + half * 2;
      v2f b;
      b.x = trow[(c0 + 0) * WF + n0 + l];
      b.y = trow[(c0 + 1) * WF + n0 + l];
      acc = __builtin_amdgcn_wmma_f32_16x16x4_f32(
          /*neg_a=*/false, afr[ks], /*neg_b=*/false, b,
          /*c_mod=*/(short)0, acc, /*reuse_a=*/false, /*reuse_b=*/false);
    }
    // C/D layout: VGPR r -> M=r (lanes 0-15) / M=8+r (lanes 16-31), N = lane%16
#pragma unroll
    for (int r = 0; r < 8; ++r) {
      const int row = r + half * 8;
      M[(m0 + row) * LDST + (n0 + l)] = acc[r];
    }
  }
  __syncthreads();

  // Phase 2: cost[d][h][w] = (w >= d) ? M[w][w-d] : 0   (coalesced global stores)
  for (int idx = threadIdx.x; idx < WF * DFULL; idx += 384) {
    const int d = idx / WF;
    const int w = idx - d * WF;
    const float v = (w >= d) ? M[w * LDST + (w - d)] : 0.0f;
    cost[(size_t)d * (HF * WF) + (size_t)h * WF + w] = v;
  }
}

// ---------------------------------------------------------------------------
// Kernel 3: trilinear upsample (48->192 D, 96->384 H, 192->768 W) + argmax + clamp
// Rolling cLo/cHi so each of the 48 planes is bilinearly fetched exactly once.
// ---------------------------------------------------------------------------
__global__ __launch_bounds__(256) void upsample_argmax(const float* __restrict__ cost,
                                                       float* __restrict__ pred) {
  int pix = blockIdx.x * blockDim.x + threadIdx.x;
  if (pix >= HO * WO) return;
  const int ho = pix / WO;
  const int wo = pix - ho * WO;

  float hs = fminf(fmaxf((ho + 0.5f) * 0.25f - 0.5f, 0.0f), (float)(HF - 1));
  float ws = fminf(fmaxf((wo + 0.5f) * 0.25f - 0.5f, 0.0f), (float)(WF - 1));
  const int h0 = (int)hs;
  const int h1 = min(h0 + 1, HF - 1);
  const float fh = hs - (float)h0;
  const int w0 = (int)ws;
  const int w1 = min(w0 + 1, WF - 1);
  const float fw = ws - (float)w0;

  const int i00 = h0 * WF + w0, i01 = h0 * WF + w1;
  const int i10 = h1 * WF + w0, i11 = h1 * WF + w1;
  const float q00 = (1.f - fh) * (1.f - fw), q01 = (1.f - fh) * fw;
  const float q10 = fh * (1.f - fw),         q11 = fh * fw;

  auto fetch = [&](int k) -> float {
    const float* __restrict__ p = cost + (size_t)k * (HF * WF);
    return p[i00] * q00 + p[i01] * q01 + p[i10] * q10 + p[i11] * q11;
  };

  int curK = 0;
  float cLo = fetch(0);
  float cHi = fetch(1);
  float best = -3.402823466e38f;
  int bestd = 0;
  for (int d = 0; d < DFULL; ++d) {
    float ds = (d + 0.5f) * 0.25f - 0.5f;
    ds = fminf(fmaxf(ds, 0.0f), (float)(DC - 1));
    int k0 = (int)ds;
    if (k0 > DC - 2) k0 = DC - 2;
    const float t = ds - (float)k0;
    if (k0 != curK) {  // k0 increases by at most 1 per step
      curK = k0;
      cLo = cHi;
      cHi = fetch(min(k0 + 1, DC - 1));
    }
    const float v = cLo + t * (cHi - cLo);
    if (v > best) { best = v; bestd = d; }  // strict > keeps first tie (jnp.argmax)
  }
  pred[pix] = fmaxf((float)bestd, 1.0f);
}

// ---------------------------------------------------------------------------
extern "C" void kernel_launch(void* const* d_in, const int* in_sizes, int n_in,
                              void* d_out, int out_size, void* d_ws, size_t ws_size,
                              hipStream_t stream) {
  const float* left_fea  = (const float*)d_in[0];
  const float* right_fea = (const float*)d_in[1];
  // d_in[2..4] = H, W, maxdisp scalars; shapes are fixed by the reference setup.

  float* rn = (float*)d_ws;                          // [96][32][192]
  float* tn = rn + (size_t)HF * CF * WF;             // [96][32][192]

  float* cost_init = (float*)d_out;                  // [192][96][192]
  float* pred = cost_init + (size_t)DFULL * HF * WF; // [384][768]

  {
    const int n = HF * WF;
    normalize_feats<<<(n + 255) / 256, 256, 0, stream>>>(left_fea, right_fea, rn, tn);
  }
  corr_rows<<<HF, 384, LDS_BYTES, stream>>>(rn, tn, cost_init);
  {
    const int n = HO * WO;
    upsample_argmax<<<(n + 255) / 256, 256, 0, stream>>>(cost_init, pred);
  }
}